// StateMixer_42623255445875
// MI455X (gfx1250) — compile-verified
//
#include <hip/hip_runtime.h>
#include <cstddef>

// ---------------------------------------------------------------------------
// Problem constants (from the reference)
// ---------------------------------------------------------------------------
#define T_   3
#define N_   200000
#define C_   256          // CIN == C == 256
#define B_   512
#define G_   64
#define HC_  (G_ + 3 * C_) // 832
#define H_   512
#define GG_  256
#define EPS_ 1e-5f

typedef float v2f __attribute__((ext_vector_type(2)));
typedef float v8f __attribute__((ext_vector_type(8)));

__device__ __forceinline__ v8f v8f_zero() {
  v8f z = {0.f, 0.f, 0.f, 0.f, 0.f, 0.f, 0.f, 0.f};
  return z;
}

__device__ __forceinline__ int lower_bound_i(const int* a, int n, int key) {
  int lo = 0, hi = n;
  while (lo < hi) {
    int mid = (lo + hi) >> 1;
    if (a[mid] < key) lo = mid + 1; else hi = mid;
  }
  return lo;
}

// ---------------------------------------------------------------------------
// Kernel 1: xr[t][c] = sum_k gtok[t][k] * Wr[t][k][c] + br[t][c]   (tiny)
// ---------------------------------------------------------------------------
__global__ void xr_kernel(const float* __restrict__ gtok,
                          const float* __restrict__ Wr,
                          const float* __restrict__ br,
                          float* __restrict__ xr) {
  const int t = blockIdx.x;
  const int c = threadIdx.x;
  const float* Wt = Wr + (size_t)t * C_ * C_;
  const float* gt = gtok + t * C_;
  float acc = br[t * C_ + c];
  for (int k = 0; k < C_; ++k) acc += gt[k] * Wt[(size_t)k * C_ + c];
  xr[t * C_ + c] = acc;
}

// ---------------------------------------------------------------------------
// Kernel 2: fused GATv2 per (t, graph b).
//   One workgroup (256 threads = 8 wave32) owns one contiguous node segment.
//   Streams 32x256 tiles of x through LDS, computes xl tiles with fp32 WMMA,
//   and does an online (flash-style) segment softmax + weighted aggregation.
// ---------------------------------------------------------------------------
__global__ __launch_bounds__(256) void gat_fused_kernel(
    const float* __restrict__ x, const float* __restrict__ Wl,
    const float* __restrict__ bl, const float* __restrict__ att,
    const float* __restrict__ bias_out, const float* __restrict__ xr,
    const int* __restrict__ batch, const float* __restrict__ global_attr,
    float* __restrict__ g_out, float* __restrict__ hcat) {
  const int b   = blockIdx.x;
  const int t   = blockIdx.y;
  const int tid = threadIdx.x;
  const int lane = tid & 31;      // wave32
  const int wv   = tid >> 5;      // wave id 0..7
  const int rsel = lane >> 4;     // 0/1: K-pair select (ISA A/B layout)
  const int nl   = lane & 15;

  __shared__ float xs[32][258];   // 32-row x tile, padded (258 % 64 == 2 banks)
  __shared__ float e_sh[32];
  __shared__ float w_sh[32];
  __shared__ float G_sh[C_];
  __shared__ float m_run, d_run, scale_sh;

  const int* brow = batch + (size_t)t * N_;
  const int start = lower_bound_i(brow, N_, b);
  const int end   = lower_bound_i(brow, N_, b + 1);

  // static slice of hcat: global_attr
  if (t == 0 && tid < G_)
    hcat[(size_t)b * HC_ + tid] = global_attr[(size_t)b * G_ + tid];

  if (start >= end) {  // empty graph: g = bias_out (reference den>0 guard)
    float gv = bias_out[t * C_ + tid];
    g_out[((size_t)t * B_ + b) * C_ + tid] = gv;
    hcat[(size_t)b * HC_ + G_ + t * C_ + tid] = gv;
    return;
  }

  G_sh[tid] = 0.f;
  if (tid == 0) { m_run = -__builtin_inff(); d_run = 0.f; }

  // wave tiling inside the 32x256 tile: 2 row-groups x 4 col-groups
  const int m0 = (wv & 1) * 16;
  const int n0 = (wv >> 1) * 64;

  int   colv[4];
  float blc[4], xrc[4], attc[4];
#pragma unroll
  for (int j = 0; j < 4; ++j) {
    const int col = n0 + j * 16 + nl;
    colv[j] = col;
    blc[j]  = bl[t * C_ + col];
    xrc[j]  = xr[t * C_ + col];
    attc[j] = att[t * C_ + col];
  }
  const float* Wt = Wl + (size_t)t * C_ * C_;
  __syncthreads();

  for (int r0 = start; r0 < end; r0 += 32) {
    const int vr = min(32, end - r0);

    // ---- stage x tile (coalesced float4), zero-pad invalid rows ----
#pragma unroll
    for (int i = 0; i < 8; ++i) {
      const int lin = tid + i * 256;     // 0..2047 float4 slots
      const int row = lin >> 6;
      const int c4  = (lin & 63) << 2;
      float4 v = {0.f, 0.f, 0.f, 0.f};
      if (row < vr)
        v = *reinterpret_cast<const float4*>(
            x + ((size_t)t * N_ + (r0 + row)) * C_ + c4);
      xs[row][c4 + 0] = v.x; xs[row][c4 + 1] = v.y;
      xs[row][c4 + 2] = v.z; xs[row][c4 + 3] = v.w;
    }
    if (tid < 32) e_sh[tid] = 0.f;
    // prefetch next tile (speculative; global_prefetch_b8)
    {
      const int nr0 = r0 + 32;
      const int row = tid >> 3;
      if (nr0 + row < end)
        __builtin_prefetch(x + ((size_t)t * N_ + nr0 + row) * C_ +
                               ((tid & 7) << 5), 0, 1);
    }
    __syncthreads();

    // ---- fp32 WMMA GEMM: xl_tile = x_tile(32x256) @ Wl (256x256 cols n0..) -
    v8f acc[4];
#pragma unroll
    for (int j = 0; j < 4; ++j) acc[j] = v8f_zero();

    for (int k = 0; k < C_; k += 4) {
      const int kk = k + (rsel << 1);
      // A 16x4 fragment: lanes<16 -> K={k,k+1}, lanes>=16 -> K={k+2,k+3}
      v2f av = *reinterpret_cast<const v2f*>(&xs[m0 + nl][kk]);
#pragma unroll
      for (int j = 0; j < 4; ++j) {
        v2f bv;
        bv.x = Wt[(size_t)kk * C_ + colv[j]];
        bv.y = Wt[(size_t)(kk + 1) * C_ + colv[j]];
        acc[j] = __builtin_amdgcn_wmma_f32_16x16x4_f32(
            false, av, false, bv, (short)0, acc[j], false, false);
      }
    }

    // ---- edge logits e = sum_c leakyrelu(xl + xr) * att ----
    float elane[8];
    float xlv[4][8];
#pragma unroll
    for (int r = 0; r < 8; ++r) elane[r] = 0.f;
#pragma unroll
    for (int j = 0; j < 4; ++j) {
#pragma unroll
      for (int r = 0; r < 8; ++r) {
        const float v = acc[j][r] + blc[j];   // xl (row m0+r(+8), col colv[j])
        xlv[j][r] = v;
        float s = v + xrc[j];
        s = (s > 0.f) ? s : 0.2f * s;         // leaky_relu slope 0.2
        elane[r] += s * attc[j];
      }
    }
#pragma unroll
    for (int r = 0; r < 8; ++r)
      atomicAdd(&e_sh[m0 + (rsel << 3) + r], elane[r]);  // ds_add_f32
    __syncthreads();

    // ---- online softmax update (wave 0) ----
    if (tid < 32) {
      float e = (tid < vr) ? e_sh[tid] : -__builtin_inff();
      float em = e;
      for (int off = 16; off >= 1; off >>= 1)
        em = fmaxf(em, __shfl_xor(em, off, 32));
      const float newm = fmaxf(m_run, em);
      const float wvv = (tid < vr) ? expf(e - newm) : 0.f;
      w_sh[tid] = wvv;
      float ds = wvv;
      for (int off = 16; off >= 1; off >>= 1) ds += __shfl_xor(ds, off, 32);
      if (tid == 0) {
        const float sc = expf(m_run - newm);  // first tile: exp(-inf)=0
        scale_sh = sc;
        d_run = d_run * sc + ds;
        m_run = newm;
      }
    }
    __syncthreads();
    G_sh[tid] *= scale_sh;
    __syncthreads();

    // ---- weighted accumulation G[col] += sum_r w[row] * xl[row][col] ----
#pragma unroll
    for (int j = 0; j < 4; ++j) {
      float p = 0.f;
#pragma unroll
      for (int r = 0; r < 8; ++r)
        p += w_sh[m0 + (rsel << 3) + r] * xlv[j][r];
      atomicAdd(&G_sh[colv[j]], p);
    }
    __syncthreads();
  }

  const float den = d_run;
  const float gv = G_sh[tid] / (den > 0.f ? den : 1.f) + bias_out[t * C_ + tid];
  g_out[((size_t)t * B_ + b) * C_ + tid] = gv;
  hcat[(size_t)b * HC_ + G_ + t * C_ + tid] = gv;
}

// ---------------------------------------------------------------------------
// Kernel 3: generic fp32 WMMA GEMM for the mixer.
//   out[M,Nc] = A[M,K] @ W[K,Nc] + bias   (row-major)
//   flags: bit0 = tanh epilogue, bit1 = accumulate into out (out += ...)
//   Requires M,Nc multiples of 64, K multiple of 4. Grid = (M/64, Nc/64).
// ---------------------------------------------------------------------------
__global__ __launch_bounds__(256) void gemm_f32_wmma(
    const float* __restrict__ A, const float* __restrict__ W,
    const float* __restrict__ bias, float* __restrict__ out,
    int M, int K, int Nc, int flags) {
  (void)M;
  const int tid  = threadIdx.x;
  const int lane = tid & 31;
  const int wv   = tid >> 5;
  const int rsel = lane >> 4;
  const int nl   = lane & 15;

  const int m0 = blockIdx.x * 64 + (wv & 3) * 16;   // 4 row-groups
  const int n0 = blockIdx.y * 64 + (wv >> 2) * 32;  // 2 col-groups x 2 tiles
  const int col0 = n0 + nl;
  const int col1 = n0 + 16 + nl;
  const int arow = m0 + nl;

  v8f acc0 = v8f_zero(), acc1 = v8f_zero();
  for (int k = 0; k < K; k += 4) {
    const int kk = k + (rsel << 1);
    v2f av = *reinterpret_cast<const v2f*>(A + (size_t)arow * K + kk);
    v2f b0; b0.x = W[(size_t)kk * Nc + col0]; b0.y = W[(size_t)(kk + 1) * Nc + col0];
    v2f b1; b1.x = W[(size_t)kk * Nc + col1]; b1.y = W[(size_t)(kk + 1) * Nc + col1];
    acc0 = __builtin_amdgcn_wmma_f32_16x16x4_f32(false, av, false, b0,
                                                 (short)0, acc0, false, false);
    acc1 = __builtin_amdgcn_wmma_f32_16x16x4_f32(false, av, false, b1,
                                                 (short)0, acc1, false, false);
  }

  const float bi0 = bias[col0], bi1 = bias[col1];
#pragma unroll
  for (int r = 0; r < 8; ++r) {
    const int row = m0 + (rsel << 3) + r;
    float v0 = acc0[r] + bi0;
    float v1 = acc1[r] + bi1;
    if (flags & 1) { v0 = tanhf(v0); v1 = tanhf(v1); }
    const size_t o0 = (size_t)row * Nc + col0;
    const size_t o1 = (size_t)row * Nc + col1;
    if (flags & 2) { out[o0] += v0; out[o1] += v1; }
    else           { out[o0]  = v0; out[o1]  = v1; }
  }
}

// ---------------------------------------------------------------------------
// Kernel 4: training-mode BatchNorm1d (biased var over 512 rows) + tanh.
//   One block per channel; 512 rows, 2 per thread.
// ---------------------------------------------------------------------------
__global__ __launch_bounds__(256) void bn_tanh_kernel(
    const float* __restrict__ in, const float* __restrict__ gam,
    const float* __restrict__ bet, float* __restrict__ out, int Nc) {
  const int c = blockIdx.x;
  const int tid = threadIdx.x;
  __shared__ float s1[256], s2[256];
  const float v0 = in[(size_t)tid * Nc + c];
  const float v1 = in[(size_t)(tid + 256) * Nc + c];
  s1[tid] = v0 + v1;
  s2[tid] = v0 * v0 + v1 * v1;
  __syncthreads();
  for (int off = 128; off >= 1; off >>= 1) {
    if (tid < off) { s1[tid] += s1[tid + off]; s2[tid] += s2[tid + off]; }
    __syncthreads();
  }
  const float mu  = s1[0] * (1.f / 512.f);
  const float var = fmaxf(s2[0] * (1.f / 512.f) - mu * mu, 0.f);
  const float inv = rsqrtf(var + EPS_);
  const float ga = gam[c], be = bet[c];
  out[(size_t)tid * Nc + c]         = tanhf((v0 - mu) * inv * ga + be);
  out[(size_t)(tid + 256) * Nc + c] = tanhf((v1 - mu) * inv * ga + be);
}

// ---------------------------------------------------------------------------
// Host side
// ---------------------------------------------------------------------------
extern "C" void kernel_launch(void* const* d_in, const int* in_sizes, int n_in,
                              void* d_out, int out_size, void* d_ws,
                              size_t ws_size, hipStream_t stream) {
  (void)in_sizes; (void)n_in; (void)out_size; (void)ws_size;
  const float* x           = (const float*)d_in[0];
  const float* global_attr = (const float*)d_in[1];
  const float* Wl          = (const float*)d_in[2];
  const float* bl          = (const float*)d_in[3];
  const float* Wr          = (const float*)d_in[4];
  const float* br          = (const float*)d_in[5];
  const float* att         = (const float*)d_in[6];
  const float* bias_out    = (const float*)d_in[7];
  const float* gtok        = (const float*)d_in[8];
  const float* mix1_pw = (const float*)d_in[9];
  const float* mix1_pb = (const float*)d_in[10];
  const float* mix1_w1 = (const float*)d_in[11];
  const float* mix1_b1 = (const float*)d_in[12];
  const float* mix1_w2 = (const float*)d_in[13];
  const float* mix1_b2 = (const float*)d_in[14];
  const float* bn1_g   = (const float*)d_in[15];
  const float* bn1_b   = (const float*)d_in[16];
  const float* mix2_w1 = (const float*)d_in[17];
  const float* mix2_b1 = (const float*)d_in[18];
  const float* mix2_w2 = (const float*)d_in[19];
  const float* mix2_b2 = (const float*)d_in[20];
  const float* bn2_g   = (const float*)d_in[21];
  const float* bn2_b   = (const float*)d_in[22];
  const float* mix3_pw = (const float*)d_in[23];
  const float* mix3_pb = (const float*)d_in[24];
  const float* mix3_w1 = (const float*)d_in[25];
  const float* mix3_b1 = (const float*)d_in[26];
  const float* mix3_w2 = (const float*)d_in[27];
  const float* mix3_b2 = (const float*)d_in[28];
  const int*   batch   = (const int*)d_in[29];

  // workspace layout (floats); total ~6.4 MB
  float* ws   = (float*)d_ws;
  float* xr   = ws;                          // T_*C_ (padded to 1024)
  float* hcat = ws + 1024;                   // B_*HC_
  float* T1   = hcat + (size_t)B_ * HC_;     // B_*H_
  float* Hb1  = T1 + (size_t)B_ * H_;        // B_*H_
  float* Hb2  = Hb1 + (size_t)B_ * H_;       // B_*H_
  float* Hb3  = Hb2 + (size_t)B_ * H_;       // B_*H_
  float* T3   = Hb3 + (size_t)B_ * H_;       // B_*GG_

  float* g_out = (float*)d_out;                  // [T,B,C]
  float* h_out = g_out + (size_t)T_ * B_ * C_;   // [B,GG]

  // GATv2 (dominant): dest-token transform, then fused GEMM+softmax+aggregate
  xr_kernel<<<dim3(T_), dim3(C_), 0, stream>>>(gtok, Wr, br, xr);
  gat_fused_kernel<<<dim3(B_, T_), dim3(256), 0, stream>>>(
      x, Wl, bl, att, bias_out, xr, batch, global_attr, g_out, hcat);

  // Mixer: ResidualLinear(832->512) + BN + tanh
  gemm_f32_wmma<<<dim3(8, 8), 256, 0, stream>>>(hcat, mix1_w1, mix1_b1, T1, 512, HC_, H_, 1);
  gemm_f32_wmma<<<dim3(8, 8), 256, 0, stream>>>(hcat, mix1_pw, mix1_pb, Hb1, 512, HC_, H_, 0);
  gemm_f32_wmma<<<dim3(8, 8), 256, 0, stream>>>(T1, mix1_w2, mix1_b2, Hb1, 512, H_, H_, 2);
  bn_tanh_kernel<<<dim3(H_), 256, 0, stream>>>(Hb1, bn1_g, bn1_b, Hb2, H_);

  // ResidualLinear(512->512, identity) + BN + tanh
  gemm_f32_wmma<<<dim3(8, 8), 256, 0, stream>>>(Hb2, mix2_w1, mix2_b1, T1, 512, H_, H_, 1);
  hipMemcpyAsync(Hb3, Hb2, (size_t)B_ * H_ * sizeof(float),
                 hipMemcpyDeviceToDevice, stream);
  gemm_f32_wmma<<<dim3(8, 8), 256, 0, stream>>>(T1, mix2_w2, mix2_b2, Hb3, 512, H_, H_, 2);
  bn_tanh_kernel<<<dim3(H_), 256, 0, stream>>>(Hb3, bn2_g, bn2_b, Hb1, H_);

  // ResidualLinear(512->256) -> h output
  gemm_f32_wmma<<<dim3(8, 4), 256, 0, stream>>>(Hb1, mix3_w1, mix3_b1, T3, 512, H_, GG_, 1);
  gemm_f32_wmma<<<dim3(8, 4), 256, 0, stream>>>(Hb1, mix3_pw, mix3_pb, h_out, 512, H_, GG_, 0);
  gemm_f32_wmma<<<dim3(8, 4), 256, 0, stream>>>(T3, mix3_w2, mix3_b2, h_out, 512, GG_, GG_, 2);
}